// Q_1In1Out_Conv1D_49770081026305
// MI455X (gfx1250) — compile-verified
//
#include <hip/hip_runtime.h>
#include <cstdint>
#include <cstddef>

// Problem geometry (reference: x (128, 8192) f32, k=4, out (128, 1, 8189) f32)
#define N_COLS    8192
#define KSZ       4
#define L_OUT     (N_COLS - KSZ + 1)      // 8189
#define KHALO     (KSZ - 1)               // 3
#define TILE_OUT  1024
#define TILE_LOAD (TILE_OUT + KHALO)      // 1027 elements staged in LDS
#define BLOCK     256
#define PT        (TILE_OUT / BLOCK)      // 4 outputs per thread
#define WIN       (PT + KHALO)            // 7-element register window

#if defined(__HIP_DEVICE_COMPILE__) && defined(__gfx1250__) && \
    __has_builtin(__builtin_amdgcn_tensor_load_to_lds)
#define USE_TDM 1
#else
#define USE_TDM 0
#endif

typedef __attribute__((ext_vector_type(4))) unsigned int v4u;
typedef __attribute__((ext_vector_type(8))) int v8i;
typedef __attribute__((ext_vector_type(4))) int v4i;

__global__ __launch_bounds__(BLOCK)
void qconv4_kernel(const float* __restrict__ x,
                   const float* __restrict__ w,
                   float* __restrict__ out)
{
    __shared__ float lv[TILE_LOAD + 1];   // +1 pad

    const int row   = blockIdx.y;
    const int start = blockIdx.x * TILE_OUT;
    int nload = N_COLS - start;           // clamp at row edge: DMA never reads OOB
    if (nload > TILE_LOAD) nload = TILE_LOAD;
    const float* gsrc = x + (size_t)row * N_COLS + start;

#if USE_TDM
    // --- CDNA5 Tensor Data Mover: DMA the row segment Global -> LDS ---------
    if (threadIdx.x == 0) {               // one wave issues (EXEC ignored by TDM)
        const unsigned long long ga = (unsigned long long)(uintptr_t)gsrc;
        const unsigned int lds = (unsigned int)(uintptr_t)&lv[0];

        // D# group 0: count=1 | lds_addr[63:32] | global_addr[120:64] | type=2
        v4u g0;
        g0.x = 1u;
        g0.y = lds;
        g0.z = (unsigned int)(ga & 0xffffffffull);
        g0.w = (unsigned int)((ga >> 32) & 0x01ffffffull) | (2u << 30);

        // D# group 1: data_size=2 (4B elems), 1-row tile of nload elements
        v8i g1;
        g1[0] = 0x00020000;                                  // data_size=4B
        g1[1] = (int)(((unsigned)nload & 0xffffu) << 16);    // tensor_dim0[15:0] @ [63:48]
        g1[2] = (int)(1u << 16);                             // tensor_dim0[31:16]=0, tensor_dim1=1
        g1[3] = (int)(((unsigned)nload & 0xffffu) << 16);    // tile_dim0 @ [127:112]
        g1[4] = 1;                                           // tile_dim1=1, tile_dim2=0
        g1[5] = N_COLS;                                      // tensor_dim0_stride
        g1[6] = 0;
        g1[7] = 0;

        v4i gz = {0, 0, 0, 0};
#if __clang_major__ >= 23
        v8i gz8 = {0, 0, 0, 0, 0, 0, 0, 0};
        __builtin_amdgcn_tensor_load_to_lds(g0, g1, gz, gz, gz8, 0);
#else
        __builtin_amdgcn_tensor_load_to_lds(g0, g1, gz, gz, 0);
#endif
        __builtin_amdgcn_s_wait_tensorcnt(0);                // s_wait_tensorcnt 0
    }
#else
    // Fallback (host pass / builtin unavailable): cooperative coalesced fill
    for (int i = threadIdx.x; i < nload; i += BLOCK) lv[i] = gsrc[i];
#endif
    __syncthreads();

    // --- closed-form circuit coefficients (omega = w[2] provably drops out) --
    // Use native v_sin_f32 / v_cos_f32: weights are N(0,1) scalars feeding a
    // final multiply; the libm Payne-Hanek path costs ~650 instructions here.
    const float phi   = w[0];
    const float theta = w[1];
#if defined(__HIP_DEVICE_COMPILE__)
    const float sT = __sinf(theta);
    const float C0 = __cosf(theta);        //  coeff of z1*z2*z3
    const float C1 = -sT * __cosf(phi);    //  coeff of x0*x1
    const float C2 =  sT * __sinf(phi);    //  coeff of x0*y1*z2*z3
#else
    const float sT = sinf(theta);
    const float C0 = cosf(theta);
    const float C1 = -sT * cosf(phi);
    const float C2 =  sT * sinf(phi);
#endif

    // --- per-element Bloch features over a 7-wide register window -----------
    // z = 1/sqrt(1+v^2); x = v * z / sqrt(1+v^4); y = v^2 * x
    const int base = threadIdx.x * PT;
    float zf[WIN], xf[WIN], yf[WIN];
#pragma unroll
    for (int j = 0; j < WIN; ++j) {
        const float v  = lv[base + j];
        const float v2 = v * v;
        const float r  = __builtin_amdgcn_rsqf(1.0f + v2);        // v_rsq_f32
        const float q  = __builtin_amdgcn_rsqf(1.0f + v2 * v2);   // v_rsq_f32
        const float xx = v * r * q;
        zf[j] = r;
        xf[j] = xx;
        yf[j] = v2 * xx;
    }

    // --- combine: out[l] = z2*z3*(C0*z1 + C2*x0*y1) + C1*x0*x1 --------------
    float* orow = out + (size_t)row * L_OUT;
#pragma unroll
    for (int i = 0; i < PT; ++i) {
        const int l = start + base + i;
        if (l < L_OUT) {
            const float zz = zf[i + 2] * zf[i + 3];
            const float t  = fmaf(C2 * xf[i], yf[i + 1], C0 * zf[i + 1]);
            orow[l] = fmaf(zz, t, C1 * xf[i] * xf[i + 1]);
        }
    }
}

extern "C" void kernel_launch(void* const* d_in, const int* in_sizes, int n_in,
                              void* d_out, int out_size, void* d_ws, size_t ws_size,
                              hipStream_t stream) {
    const float* x = (const float*)d_in[0];     // (B, 8192) f32
    const float* w = (const float*)d_in[1];     // (3,) f32
    float* out = (float*)d_out;                 // (B, 1, 8189) f32
    const int rows = in_sizes[0] / N_COLS;      // 128
    dim3 grid((L_OUT + TILE_OUT - 1) / TILE_OUT, rows, 1);  // (8, 128)
    dim3 block(BLOCK, 1, 1);
    qconv4_kernel<<<grid, block, 0, stream>>>(x, w, out);
}